// WaveNet_87900800680405
// MI455X (gfx1250) — compile-verified
//
#include <hip/hip_runtime.h>

typedef __attribute__((ext_vector_type(2))) float v2f;
typedef __attribute__((ext_vector_type(8))) float v8f;

#define N_RES   20
#define NPOS    96                    // time positions computed per batch
#define LPAD    2                     // zero slots for the d=2 left pad (t = -1)
#define RPAD    6                     // zero slots for reads at t+1 past the tile region
#define TOT     (LPAD + NPOS + RPAD)  // 104 float2 slots per buffer
#define NTILE   6                     // 6 * 16 = 96 positions
#define OUT_T   64
#define SEQ_L   32768

__global__ __launch_bounds__(32)
void wavenet_wmma_kernel(const float* __restrict__ x,
                         const float* __restrict__ ciw,
                         const float* __restrict__ cib,
                         const float* __restrict__ wd,
                         const float* __restrict__ bd,
                         const float* __restrict__ wg,
                         const float* __restrict__ bg,
                         const float* __restrict__ cow,
                         const float* __restrict__ cob,
                         float* __restrict__ out)
{
    __shared__ float2 hbuf[2][TOT];          // ping-pong h (2 filter channels packed)
    __shared__ float  lwd[N_RES * 8], lwg[N_RES * 8];
    __shared__ float  lbd[N_RES * 2], lbg[N_RES * 2];

    const int lane = threadIdx.x;            // 0..31 (wave32)
    const int bidx = blockIdx.x;             // batch element
    const int n    = lane & 15;              // N (time within tile) / M (A row)
    const int kh   = lane >> 4;              // lane half selects K pair {2kh, 2kh+1}

    // Stage all layer weights/biases to LDS once (coalesced, tiny).
    for (int j = lane; j < N_RES * 8; j += 32) { lwd[j] = wd[j]; lwg[j] = wg[j]; }
    for (int j = lane; j < N_RES * 2; j += 32) { lbd[j] = bd[j]; lbg[j] = bg[j]; }

    // conv_in (1x1): h0[t] = ciw[f]*x[t] + cib[f]; only t in [0,95] is live.
    {
        const float w0 = ciw[0], w1 = ciw[1];
        const float c0 = cib[0], c1 = cib[1];
        const float* xb = x + (size_t)bidx * SEQ_L;
        for (int t = lane; t < NPOS; t += 32) {
            float xv = xb[t];
            hbuf[0][LPAD + t] = make_float2(fmaf(w0, xv, c0), fmaf(w1, xv, c1));
        }
        for (int j = lane; j < TOT; j += 32) {
            if (j < LPAD || j >= LPAD + NPOS) hbuf[0][j] = make_float2(0.f, 0.f);
            hbuf[1][j] = make_float2(0.f, 0.f);   // pads stay zero forever
        }
    }
    __syncthreads();

    float2 skip[4];                           // skip sums for t = 0..63 (tiles 0..3)
    skip[0] = skip[1] = skip[2] = skip[3] = make_float2(0.f, 0.f);

    int cur = 0;
    for (int i = 0; i < N_RES; ++i) {
        const int dil = (i < 10) ? 1 : 2;     // dilation: 2^(i/10)
        const int o0  = (dil == 1) ? 0 : -1;  // tap-0 offset (d=2 has pad=1 -> t-1)
        const int rsh = (dil == 1) ? 1 : 0;   // residual source offset

        // A (16x4): rows 0-3 = [cd0 cd1 cg0 cg1] x [h0@tap0 h1@tap0 h0@tap1 h1@tap1]
        // layout: lane m = n, K = 2*kh + {0,1} across the 2 VGPRs.
        v2f a; a.x = 0.f; a.y = 0.f;
        if (n < 4) {
            const float* wsrc = (n < 2) ? lwd : lwg;
            const int fout = n & 1;
            const int kb   = kh * 2;
            // wd[i][fout][fin][ktap], k = fin + 2*ktap
            a.x = wsrc[i * 8 + fout * 4 + ((kb    ) & 1) * 2 + ((kb    ) >> 1)];
            a.y = wsrc[i * 8 + fout * 4 + ((kb + 1) & 1) * 2 + ((kb + 1) >> 1)];
        }

        // C = bias broadcast across N (rows 0-3 live in lanes 0-15 of c[0..3]).
        v8f cacc = {0.f, 0.f, 0.f, 0.f, 0.f, 0.f, 0.f, 0.f};
        if (kh == 0) {
            cacc[0] = lbd[i * 2 + 0];
            cacc[1] = lbd[i * 2 + 1];
            cacc[2] = lbg[i * 2 + 0];
            cacc[3] = lbg[i * 2 + 1];
        }

        const int rdo = kh ? 1 : o0;          // B read offset per lane half

        // Phase 1: batch all LDS loads for this layer (no store aliasing in between,
        // so they issue back-to-back and the dscnt waits pipeline).
        float2 bvv[NTILE], rvv[NTILE];
#pragma unroll
        for (int tj = 0; tj < NTILE; ++tj) {
            const int t = tj * 16 + n;
            bvv[tj] = hbuf[cur][LPAD + t + rdo];   // B column (K pair for this half)
            rvv[tj] = hbuf[cur][LPAD + t + rsh];   // residual source
        }

        // Phase 2: 6 WMMAs + gating. Lanes 16-31 produce exact zeros for g (A rows
        // 8-11 and C rows 8-11 are zero), so no EXEC predication is needed here.
        float2 hout[NTILE];
#pragma unroll
        for (int tj = 0; tj < NTILE; ++tj) {
            v2f bb; bb.x = bvv[tj].x; bb.y = bvv[tj].y;
            v8f dm = __builtin_amdgcn_wmma_f32_16x16x4_f32(
                false, a, false, bb, (short)0, cacc, false, false);
            // lane n (half 0): dm[0]=cd0, dm[1]=cd1, dm[2]=cg0, dm[3]=cg1 at position t
            float g0 = dm[0] * __builtin_amdgcn_rcpf(1.0f + __expf(-dm[2]));
            float g1 = dm[1] * __builtin_amdgcn_rcpf(1.0f + __expf(-dm[3]));
            if (tj < 4) { skip[tj].x += g0; skip[tj].y += g1; }   // upper half adds 0
            hout[tj] = make_float2(rvv[tj].x + g0, rvv[tj].y + g1);
        }

        // Phase 3: write next-layer h (lanes 0-15 own the 16 positions of each tile).
        if (kh == 0) {
#pragma unroll
            for (int tj = 0; tj < NTILE; ++tj) {
                hbuf[cur ^ 1][LPAD + tj * 16 + n] = hout[tj];
            }
        }
        __syncthreads();   // 1-wave workgroup: lowers to S_NOP, keeps DS ordering honest
        cur ^= 1;
    }

    // conv_out (1x1) on relu(skip sum), first 64 time steps only.
    const float cw0 = cow[0], cw1 = cow[1], cb0 = cob[0];
    if (kh == 0) {
#pragma unroll
        for (int tj = 0; tj < 4; ++tj) {
            const int t = tj * 16 + n;
            float s0 = fmaxf(skip[tj].x, 0.f);
            float s1 = fmaxf(skip[tj].y, 0.f);
            out[bidx * OUT_T + t] = fmaf(cw0, s0, fmaf(cw1, s1, cb0));
        }
    }
}

extern "C" void kernel_launch(void* const* d_in, const int* in_sizes, int n_in,
                              void* d_out, int out_size, void* d_ws, size_t ws_size,
                              hipStream_t stream) {
    (void)in_sizes; (void)n_in; (void)out_size; (void)d_ws; (void)ws_size;
    wavenet_wmma_kernel<<<64, 32, 0, stream>>>(
        (const float*)d_in[0],   // x            [64,1,32768]
        (const float*)d_in[1],   // conv_in_w    [2,1,1]
        (const float*)d_in[2],   // conv_in_b    [2]
        (const float*)d_in[3],   // wd           [20,2,2,2]
        (const float*)d_in[4],   // bd           [20,2]
        (const float*)d_in[5],   // wg           [20,2,2,2]
        (const float*)d_in[6],   // bg           [20,2]
        (const float*)d_in[7],   // conv_out_w   [1,2,1]
        (const float*)d_in[8],   // conv_out_b   [1]
        (float*)d_out);          // out          [64,1,64]
}